// GraphProjector_52252572123391
// MI455X (gfx1250) — compile-verified
//
#include <hip/hip_runtime.h>
#include <hip/hip_bf16.h>
#include <stdint.h>

// Problem constants (from reference): x (4,64,64,128) -> B=4, N=4096, F=128
#define NN  4096
#define FF  128
#define EE  16777216u          // N*N
#define RANK0 15099493u        // floor(0.9 * (E-1)); frac = 0.5 exactly
#define QFRAC 0.5f

typedef __attribute__((ext_vector_type(2)))  float    v2f;
typedef __attribute__((ext_vector_type(8)))  float    v8f;

#if __has_builtin(__builtin_amdgcn_global_load_async_to_lds_b128) && \
    __has_builtin(__builtin_amdgcn_s_wait_asynccnt)
#define GP_ASYNC 1
typedef int gp_v4i __attribute__((vector_size(4 * sizeof(int))));
typedef __attribute__((address_space(1))) gp_v4i* gp_g4;   // global src
typedef __attribute__((address_space(3))) gp_v4i* gp_l4;   // LDS dst
#endif

struct SelState {
  unsigned bin1[2];
  unsigned rank1[2];
  unsigned bin2[2];
  unsigned rank2[2];
  unsigned value[2];
  float    thr;
  unsigned num_edges;
};

__device__ __forceinline__ float gp_wave_sum(float v) {
  #pragma unroll
  for (int off = 16; off; off >>= 1) v += __shfl_xor(v, off, 32);
  return v;
}

// ---------------- 1) Projections: Q[i,b] = 0.25 * x[b,i,:].wq ; K[i,b] = x[b,i,:].wk
__global__ void gp_proj(const float* __restrict__ x, const float* __restrict__ wk,
                        const float* __restrict__ wq, float* __restrict__ Qd,
                        float* __restrict__ Kd) {
  int wid  = (blockIdx.x * blockDim.x + threadIdx.x) >> 5;  // 0..16383
  int lane = threadIdx.x & 31;
  int i = wid & (NN - 1);
  int b = wid >> 12;
  const float* xr = x + ((size_t)b * NN + i) * FF;
  float q = 0.f, k = 0.f;
  #pragma unroll
  for (int t = 0; t < 4; ++t) {
    int f = lane + 32 * t;
    float xv = xr[f];
    q += xv * wq[f];
    k += xv * wk[f];
  }
  q = gp_wave_sum(q);
  k = gp_wave_sum(k);
  if (lane == 0) {
    Qd[i * 4 + b] = 0.25f * q;   // fold mean-over-batch into Q
    Kd[i * 4 + b] = k;
  }
}

// ---------------- 2) Column stats (max, first-argmax, 1/sum-exp) from rank-4 factors.
// Thread per column; Q staged in LDS (64 KB), via async-to-LDS when available.
__global__ void gp_colstats(const float* __restrict__ Qd, const float* __restrict__ Kd,
                            float* __restrict__ colmax, float* __restrict__ colinv,
                            int* __restrict__ colbest) {
  __shared__ float Qs[NN * 4];
#ifdef GP_ASYNC
  // 4096 x 16-byte async chunks; ASYNCcnt tracks completion.
  for (int ch = threadIdx.x; ch < NN; ch += blockDim.x) {
    __builtin_amdgcn_global_load_async_to_lds_b128(
        (gp_g4)(Qd + ch * 4), (gp_l4)(Qs + ch * 4), 0, 0);
  }
  __builtin_amdgcn_s_wait_asynccnt(0);
  __syncthreads();
#else
  for (int t = threadIdx.x; t < NN * 4; t += blockDim.x) Qs[t] = Qd[t];
  __syncthreads();
#endif
  int j = blockIdx.x * blockDim.x + threadIdx.x;
  float k0 = Kd[j * 4 + 0], k1 = Kd[j * 4 + 1], k2 = Kd[j * 4 + 2], k3 = Kd[j * 4 + 3];
  float mx = -3.402823466e+38f;
  int best = 0;
  for (int i = 0; i < NN; ++i) {
    float a = Qs[i * 4 + 0] * k0 + Qs[i * 4 + 1] * k1 + Qs[i * 4 + 2] * k2 + Qs[i * 4 + 3] * k3;
    if (a > mx) { mx = a; best = i; }          // strict '>' keeps first occurrence
  }
  float s = 0.f;
  for (int i = 0; i < NN; ++i) {
    float a = Qs[i * 4 + 0] * k0 + Qs[i * 4 + 1] * k1 + Qs[i * 4 + 2] * k2 + Qs[i * 4 + 3] * k3;
    s += __expf(a - mx);
  }
  colmax[j] = mx;
  colinv[j] = 1.0f / s;                        // one divide per column, not per tile
  colbest[j] = best;
}

// ---------------- 3) WMMA: one 16-row A fragment per wave, swept over 4 j-tiles.
// L = Q.K^T (K=4) via v_wmma_f32_16x16x4_f32, then P = exp(L - m_j) * colinv_j.
__global__ void gp_softmax_wmma(const float* __restrict__ Qd, const float* __restrict__ Kd,
                                const float* __restrict__ colmax, const float* __restrict__ colinv,
                                float* __restrict__ P) {
  int task = (blockIdx.x * blockDim.x + threadIdx.x) >> 5;  // 0..16383
  int lane = threadIdx.x & 31;
  int ti = task >> 6;            // 256 i-tiles
  int jg = task & 63;            // 64 groups of 4 j-tiles
  int i0 = ti << 4;
  int j0 = jg << 6;
  int m  = lane & 15;
  int kh = lane >> 4;            // which K-half this lane holds
  // 32-bit A 16x4: VGPR0 = K{0|2}, VGPR1 = K{1|3} split by lane half.
  v2f a;
  a.x = Qd[(i0 + m) * 4 + 2 * kh];
  a.y = Qd[(i0 + m) * 4 + 2 * kh + 1];
  #pragma unroll
  for (int jt = 0; jt < 4; ++jt) {
    int jb = j0 + jt * 16;
    v2f b;
    b.x = Kd[(jb + m) * 4 + 2 * kh];
    b.y = Kd[(jb + m) * 4 + 2 * kh + 1];
    v8f c = {};
    v8f d = __builtin_amdgcn_wmma_f32_16x16x4_f32(false, a, false, b, (short)0, c, false, false);
    int col = jb + m;
    float cm = colmax[col];
    float ci = colinv[col];
    #pragma unroll
    for (int v = 0; v < 8; ++v) {
      int row = i0 + v + 8 * kh; // C/D layout: vgpr v, lane-half selects M or M+8
      P[(size_t)row * NN + col] = __expf(d[v] - cm) * ci;
    }
  }
}

// ---------------- 4) Radix select (positive floats -> u32 order). Pass 1: top 12 bits.
__global__ void gp_hist1(const float* __restrict__ P, unsigned* __restrict__ hist) {
  __shared__ unsigned h[4096];
  for (int t = threadIdx.x; t < 4096; t += blockDim.x) h[t] = 0;
  __syncthreads();
  unsigned stride = blockDim.x * gridDim.x;
  for (unsigned e = blockIdx.x * blockDim.x + threadIdx.x; e < EE; e += stride) {
    unsigned u = __float_as_uint(P[e]);
    atomicAdd(&h[u >> 20], 1u);
  }
  __syncthreads();
  for (int t = threadIdx.x; t < 4096; t += blockDim.x)
    if (h[t]) atomicAdd(&hist[t], h[t]);
}

__global__ void gp_sel1(const unsigned* __restrict__ hist, SelState* st) {
  if (threadIdx.x != 0 || blockIdx.x != 0) return;
  for (int t = 0; t < 2; ++t) {
    unsigned k = RANK0 + (unsigned)t, c = 0;
    for (int bin = 0; bin < 4096; ++bin) {
      unsigned n = hist[bin];
      if (k < c + n) { st->bin1[t] = bin; st->rank1[t] = k - c; break; }
      c += n;
    }
  }
}

__global__ void gp_hist2(const float* __restrict__ P, const SelState* __restrict__ st,
                         unsigned* __restrict__ hist2) {
  __shared__ unsigned h[2][4096];
  for (int t = threadIdx.x; t < 8192; t += blockDim.x) (&h[0][0])[t] = 0;
  __syncthreads();
  unsigned b0 = st->bin1[0], b1 = st->bin1[1];
  unsigned stride = blockDim.x * gridDim.x;
  for (unsigned e = blockIdx.x * blockDim.x + threadIdx.x; e < EE; e += stride) {
    unsigned u = __float_as_uint(P[e]);
    unsigned top = u >> 20, mid = (u >> 8) & 0xFFFu;
    if (top == b0) atomicAdd(&h[0][mid], 1u);
    if (top == b1) atomicAdd(&h[1][mid], 1u);
  }
  __syncthreads();
  for (int t = threadIdx.x; t < 8192; t += blockDim.x) {
    unsigned v = (&h[0][0])[t];
    if (v) atomicAdd(&hist2[t], v);
  }
}

__global__ void gp_sel2(const unsigned* __restrict__ hist2, SelState* st) {
  if (threadIdx.x != 0 || blockIdx.x != 0) return;
  for (int t = 0; t < 2; ++t) {
    unsigned k = st->rank1[t], c = 0;
    for (int bin = 0; bin < 4096; ++bin) {
      unsigned n = hist2[t * 4096 + bin];
      if (k < c + n) { st->bin2[t] = bin; st->rank2[t] = k - c; break; }
      c += n;
    }
  }
}

__global__ void gp_hist3(const float* __restrict__ P, const SelState* __restrict__ st,
                         unsigned* __restrict__ hist3) {
  __shared__ unsigned h[2][256];
  for (int t = threadIdx.x; t < 512; t += blockDim.x) (&h[0][0])[t] = 0;
  __syncthreads();
  unsigned p0 = (st->bin1[0] << 12) | st->bin2[0];
  unsigned p1 = (st->bin1[1] << 12) | st->bin2[1];
  unsigned stride = blockDim.x * gridDim.x;
  for (unsigned e = blockIdx.x * blockDim.x + threadIdx.x; e < EE; e += stride) {
    unsigned u = __float_as_uint(P[e]);
    unsigned pre = u >> 8, lo = u & 0xFFu;
    if (pre == p0) atomicAdd(&h[0][lo], 1u);
    if (pre == p1) atomicAdd(&h[1][lo], 1u);
  }
  __syncthreads();
  for (int t = threadIdx.x; t < 512; t += blockDim.x) {
    unsigned v = (&h[0][0])[t];
    if (v) atomicAdd(&hist3[t], v);
  }
}

__global__ void gp_sel3(const unsigned* __restrict__ hist3, SelState* st) {
  if (threadIdx.x != 0 || blockIdx.x != 0) return;
  for (int t = 0; t < 2; ++t) {
    unsigned k = st->rank2[t], c = 0;
    for (int bin = 0; bin < 256; ++bin) {
      unsigned n = hist3[t * 256 + bin];
      if (k < c + n) {
        st->value[t] = (st->bin1[t] << 20) | (st->bin2[t] << 8) | (unsigned)bin;
        break;
      }
      c += n;
    }
  }
  float v0 = __uint_as_float(st->value[0]);
  float v1 = __uint_as_float(st->value[1]);
  st->thr = v0 + QFRAC * (v1 - v0);     // jnp.quantile 'linear'
}

// ---------------- 5) Per-column connection count (strict > thr); coalesced.
__global__ void gp_nconn(const float* __restrict__ P, const SelState* __restrict__ st,
                         int* __restrict__ nconn) {
  int j = blockIdx.x * blockDim.x + threadIdx.x;
  float thr = st->thr;
  int c = 0;
  for (int i = 0; i < NN; ++i) c += (P[(size_t)i * NN + j] > thr) ? 1 : 0;
  nconn[j] = c;
}

// ---------------- 6) Row norms over the FINAL mask (threshold OR argmax fix-up).
__global__ void gp_rownorm(const float* __restrict__ P, const SelState* __restrict__ st,
                           const int* __restrict__ nconn, const int* __restrict__ colbest,
                           float* __restrict__ rowNorm) {
  int wid  = (blockIdx.x * blockDim.x + threadIdx.x) >> 5;  // row
  int lane = threadIdx.x & 31;
  float thr = st->thr;
  const float* row = P + (size_t)wid * NN;
  float s = 0.f;
  for (int j = lane; j < NN; j += 32) {
    float p = row[j];
    bool m = (p > thr) || (nconn[j] == 0 && colbest[j] == wid);
    if (m) s += p;
  }
  s = gp_wave_sum(s);
  if (lane == 0) rowNorm[wid] = s;
}

// ---------------- 7) Output padding: edge_index = -1, weights = 0.
__global__ void gp_initout(int* __restrict__ oi, float* __restrict__ of) {
  size_t stride = (size_t)blockDim.x * gridDim.x;
  for (size_t t = blockIdx.x * (size_t)blockDim.x + threadIdx.x; t < 3ull * EE; t += stride) {
    if (t < 2ull * EE) oi[t] = -1;
    else               of[t] = 0.f;
  }
}

// ---------------- 8) Per-block mask counts (16384-element row-major chunks).
__global__ void gp_count(const float* __restrict__ P, const SelState* __restrict__ st,
                         const int* __restrict__ nconn, const int* __restrict__ colbest,
                         unsigned* __restrict__ blockcnt) {
  __shared__ unsigned tot;
  if (threadIdx.x == 0) tot = 0;
  __syncthreads();
  float thr = st->thr;
  unsigned base = blockIdx.x * 16384u;
  unsigned local = 0;
  for (int t = threadIdx.x; t < 16384; t += blockDim.x) {
    unsigned e = base + (unsigned)t;
    int i = e >> 12, j = e & 4095;
    float p = P[e];
    bool m = (p > thr) || (nconn[j] == 0 && colbest[j] == i);
    local += m ? 1u : 0u;
  }
  atomicAdd(&tot, local);
  __syncthreads();
  if (threadIdx.x == 0) blockcnt[blockIdx.x] = tot;
}

// ---------------- 9) Single-block exclusive scan of 1024 block counts.
__global__ void gp_scan(const unsigned* __restrict__ blockcnt, unsigned* __restrict__ blockoff,
                        SelState* st, int* __restrict__ out_ne) {
  __shared__ unsigned s[1024];
  int t = threadIdx.x;
  unsigned v = blockcnt[t];
  s[t] = v;
  __syncthreads();
  for (int off = 1; off < 1024; off <<= 1) {
    unsigned add = (t >= off) ? s[t - off] : 0u;
    __syncthreads();
    s[t] += add;
    __syncthreads();
  }
  blockoff[t] = s[t] - v;
  if (t == 1023) {
    st->num_edges = s[t];
    *out_ne = (int)s[t];
  }
}

// ---------------- 10) Ordered compaction (global row-major preserved).
__global__ void gp_write(const float* __restrict__ P, const SelState* __restrict__ st,
                         const int* __restrict__ nconn, const int* __restrict__ colbest,
                         const float* __restrict__ rowNorm, const unsigned* __restrict__ blockoff,
                         int* __restrict__ orow, int* __restrict__ ocol, float* __restrict__ ow) {
  __shared__ unsigned wsum[8];
  int tid  = threadIdx.x;
  int lane = tid & 31;
  int wv   = tid >> 5;
  float thr = st->thr;
  unsigned base   = blockoff[blockIdx.x];
  unsigned estart = blockIdx.x * 16384u;
  for (int iter = 0; iter < 64; ++iter) {
    unsigned e = estart + (unsigned)iter * 256u + (unsigned)tid;
    int i = e >> 12, j = e & 4095;
    float p = P[e];
    bool m = (p > thr) || (nconn[j] == 0 && colbest[j] == i);
    unsigned bal = __builtin_amdgcn_ballot_w32(m);
    if (lane == 0) wsum[wv] = (unsigned)__popc(bal);
    __syncthreads();
    unsigned wpre = 0, tot = 0;
    #pragma unroll
    for (int w = 0; w < 8; ++w) {
      unsigned c = wsum[w];
      wpre += (w < wv) ? c : 0u;
      tot  += c;
    }
    if (m) {
      unsigned pos = base + wpre + (unsigned)__popc(bal & ((1u << lane) - 1u));
      orow[pos] = i;
      ocol[pos] = j;
      ow[pos]   = p / rowNorm[i];
    }
    base += tot;
    __syncthreads();
  }
}

__global__ void gp_zero(unsigned* __restrict__ p, unsigned n) {
  unsigned stride = blockDim.x * gridDim.x;
  for (unsigned t = blockIdx.x * blockDim.x + threadIdx.x; t < n; t += stride) p[t] = 0u;
}

extern "C" void kernel_launch(void* const* d_in, const int* in_sizes, int n_in,
                              void* d_out, int out_size, void* d_ws, size_t ws_size,
                              hipStream_t stream) {
  (void)in_sizes; (void)n_in; (void)out_size; (void)ws_size;
  const float* x  = (const float*)d_in[0];
  const float* wk = (const float*)d_in[1];
  const float* wq = (const float*)d_in[2];

  char* ws = (char*)d_ws;
  size_t off = 0;
  float*    P        = (float*)(ws + off); off += (size_t)EE * 4;      // 64 MB
  float*    Qd       = (float*)(ws + off); off += NN * 4 * 4;
  float*    Kd       = (float*)(ws + off); off += NN * 4 * 4;
  float*    colmax   = (float*)(ws + off); off += NN * 4;
  float*    colinv   = (float*)(ws + off); off += NN * 4;
  int*      colbest  = (int*)(ws + off);   off += NN * 4;
  int*      nconn    = (int*)(ws + off);   off += NN * 4;
  float*    rowNorm  = (float*)(ws + off); off += NN * 4;
  unsigned* hist1    = (unsigned*)(ws + off); off += 4096 * 4;         // contiguous zero region start
  unsigned* hist2    = (unsigned*)(ws + off); off += 8192 * 4;
  unsigned* hist3    = (unsigned*)(ws + off); off += 512 * 4;
  SelState* st       = (SelState*)(ws + off); off += 256;              // contiguous zero region end
  unsigned* blockcnt = (unsigned*)(ws + off); off += 1024 * 4;
  unsigned* blockoff = (unsigned*)(ws + off); off += 1024 * 4;

  int*   orow = (int*)d_out;
  int*   ocol = orow + EE;
  float* ow   = (float*)d_out + 2ull * EE;
  int*   one  = (int*)d_out + 3ull * EE;

  unsigned zwords = (4096 + 8192 + 512) + 64;  // hists + state
  gp_zero        <<<64,   256, 0, stream>>>(hist1, zwords);
  gp_proj        <<<2048, 256, 0, stream>>>(x, wk, wq, Qd, Kd);
  gp_colstats    <<<16,   256, 0, stream>>>(Qd, Kd, colmax, colinv, colbest);
  gp_softmax_wmma<<<2048, 256, 0, stream>>>(Qd, Kd, colmax, colinv, P);
  gp_hist1       <<<1024, 256, 0, stream>>>(P, hist1);
  gp_sel1        <<<1,    32,  0, stream>>>(hist1, st);
  gp_hist2       <<<1024, 256, 0, stream>>>(P, st, hist2);
  gp_sel2        <<<1,    32,  0, stream>>>(hist2, st);
  gp_hist3       <<<1024, 256, 0, stream>>>(P, st, hist3);
  gp_sel3        <<<1,    32,  0, stream>>>(hist3, st);
  gp_nconn       <<<16,   256, 0, stream>>>(P, st, nconn);
  gp_rownorm     <<<512,  256, 0, stream>>>(P, st, nconn, colbest, rowNorm);
  gp_initout     <<<2048, 256, 0, stream>>>(orow, (float*)d_out);
  gp_count       <<<1024, 256, 0, stream>>>(P, st, nconn, colbest, blockcnt);
  gp_scan        <<<1,   1024, 0, stream>>>(blockcnt, blockoff, st, one);
  gp_write       <<<1024, 256, 0, stream>>>(P, st, nconn, colbest, rowNorm, blockoff, orow, ocol, ow);
}